// GNN12_19378892439972
// MI455X (gfx1250) — compile-verified
//
#include <hip/hip_runtime.h>
#include <hip/hip_bf16.h>

// ---------------------------------------------------------------------------
// MI455X (gfx1250) GNN forward. Memory-bound workload (~36 GFLOP vs ~3 GB
// traffic -> ~120us at 23.3 TB/s), so we use the native fp32 WMMA path
// (V_WMMA_F32_16X16X4_F32) for all dense GEMMs and fuse bias/relu/accumulate
// epilogues so every large tensor is touched once. Segment-sum via
// global_atomic_add_f32 (low contention: 16 edges/node avg).
// ---------------------------------------------------------------------------

typedef __attribute__((ext_vector_type(2))) float v2f;
typedef __attribute__((ext_vector_type(8))) float v8f;

#define NODES 50000
#define EDGES 800000
#define GRAPHS 50
#define EMB_D 512
#define MID_D 128
#define OUT_D 512
#define EPD_D 48

// ---- node embedding gather: x[row,:] = emb[tokens[row],:]  (512 fp32/row) --
__global__ __launch_bounds__(128) void k_embed(const int* __restrict__ tokens,
                                               const float* __restrict__ emb,
                                               float* __restrict__ x) {
  const int row = blockIdx.x;
  const int tok = tokens[row];
  const float4* s = (const float4*)(emb + (size_t)tok * EMB_D);
  float4* d = (float4*)(x + (size_t)row * EMB_D);
  d[threadIdx.x] = s[threadIdx.x];  // 128 lanes * 16B = one 2KB row
}

// ---- generic WMMA fp32 GEMM:  C[M,N] (=|+=) A[M,K] @ W[K,N] (+bias)(relu) --
// Block: 128 threads (4 waves). Block computes rows [m0,m0+16), cols
// [blockIdx.y*64, +64) with one 16x16 WMMA tile per wave.
// A-tile staged in LDS with stride K+1 (conflict-free column reads).
// V_WMMA_F32_16X16X4_F32 operand layout (ISA 7.12.2):
//   A 16x4 : lanes 0-15 -> M=lane, v[0]=K0 v[1]=K1 ; lanes 16-31 -> K2,K3
//   B 4x16 : lanes 0-15 -> N=lane, v[0]=K0 v[1]=K1 ; lanes 16-31 -> K2,K3
//   C/D    : VGPR r, lanes 0-15 -> M=r,N=lane ; lanes 16-31 -> M=r+8
__global__ __launch_bounds__(128) void k_gemm_wmma(
    const float* __restrict__ A, const float* __restrict__ W,
    const float* __restrict__ bias, float* __restrict__ C,
    int M, int N, int K, int accumulate, int relu) {
  __shared__ float As[16 * 513];  // K <= 512, stride K+1
  const int m0 = blockIdx.x * 16;
  const int tid = threadIdx.x;
  const int KP = K + 1;

  for (int r = 0; r < 16; ++r)
    for (int c = tid; c < K; c += 128)
      As[r * KP + c] = A[(size_t)(m0 + r) * K + c];
  __syncthreads();

  const int wave = tid >> 5;
  const int lane = tid & 31;
  const int half = lane >> 4;   // 0 -> K pair {0,1}, 1 -> {2,3}
  const int lr = lane & 15;
  const int n0 = blockIdx.y * 64 + wave * 16;
  if (n0 >= N) return;

  v8f acc = {0.f, 0.f, 0.f, 0.f, 0.f, 0.f, 0.f, 0.f};
  const float* As_row = &As[lr * KP + half * 2];
  const float* Wp = W + (size_t)(half * 2) * N + n0 + lr;

#pragma unroll 4
  for (int k = 0; k < K; k += 4) {
    v2f a, b;
    a.x = As_row[k];
    a.y = As_row[k + 1];
    b.x = Wp[(size_t)k * N];
    b.y = Wp[(size_t)(k + 1) * N];
    acc = __builtin_amdgcn_wmma_f32_16x16x4_f32(
        /*neg_a=*/false, a, /*neg_b=*/false, b,
        /*c_mod=*/(short)0, acc, /*reuse_a=*/false, /*reuse_b=*/false);
  }

  const float bval = bias ? bias[n0 + lr] : 0.f;
#pragma unroll
  for (int r = 0; r < 8; ++r) {
    const int row = m0 + r + half * 8;
    const size_t idx = (size_t)row * N + n0 + lr;
    float v = acc[r] + bval;
    if (accumulate) v += C[idx];
    if (relu) v = fmaxf(v, 0.f);
    C[idx] = v;
  }
}

// ---- edge message combine + segment-sum -----------------------------------
// e_raw = ms[src] + md[dst] + (ea@We + be)  ;  agg[dst] += e_raw (pre-relu,
// matching reference: relu is applied to ea only AFTER the conv).
__global__ __launch_bounds__(256) void k_edge_combine(
    const int* __restrict__ src, const int* __restrict__ dst,
    const float* __restrict__ ms, const float* __restrict__ md,
    const float* __restrict__ ew, float* __restrict__ e_out,
    float* __restrict__ agg, int relu, int store_e) {
  const int e = blockIdx.x * blockDim.x + threadIdx.x;
  if (e >= EDGES) return;
  const int s = src[e];
  const int d = dst[e];
  const float4* ps = (const float4*)(ms + (size_t)s * EPD_D);
  const float4* pd = (const float4*)(md + (size_t)d * EPD_D);
  const float4* pe = (const float4*)(ew + (size_t)e * EPD_D);
  float4* po = (float4*)(e_out + (size_t)e * EPD_D);
  float* pa = agg + (size_t)d * EPD_D;
#pragma unroll
  for (int j = 0; j < EPD_D / 4; ++j) {
    const float4 a = ps[j], b = pd[j], c = pe[j];
    float4 v;
    v.x = a.x + b.x + c.x;
    v.y = a.y + b.y + c.y;
    v.z = a.z + b.z + c.z;
    v.w = a.w + b.w + c.w;
    // segment_sum uses the raw (pre-relu) edge message
    __hip_atomic_fetch_add(pa + 4 * j + 0, v.x, __ATOMIC_RELAXED, __HIP_MEMORY_SCOPE_AGENT);
    __hip_atomic_fetch_add(pa + 4 * j + 1, v.y, __ATOMIC_RELAXED, __HIP_MEMORY_SCOPE_AGENT);
    __hip_atomic_fetch_add(pa + 4 * j + 2, v.z, __ATOMIC_RELAXED, __HIP_MEMORY_SCOPE_AGENT);
    __hip_atomic_fetch_add(pa + 4 * j + 3, v.w, __ATOMIC_RELAXED, __HIP_MEMORY_SCOPE_AGENT);
    if (store_e) {
      if (relu) {
        v.x = fmaxf(v.x, 0.f); v.y = fmaxf(v.y, 0.f);
        v.z = fmaxf(v.z, 0.f); v.w = fmaxf(v.w, 0.f);
      }
      po[j] = v;
    }
  }
}

// ---- global master pool: out[b,:] = x[first index with batch[i]==b, :] ----
__global__ __launch_bounds__(128) void k_pool(const float* __restrict__ x,
                                              const int* __restrict__ batch,
                                              float* __restrict__ out) {
  __shared__ int m;
  const int b = blockIdx.x;
  if (threadIdx.x == 0) {
    int lo = 0, hi = NODES;  // searchsorted(batch, b) on sorted batch ids
    while (lo < hi) {
      const int mid = (lo + hi) >> 1;
      if (batch[mid] < b) lo = mid + 1; else hi = mid;
    }
    m = lo;
  }
  __syncthreads();
  const float4* s = (const float4*)(x + (size_t)m * OUT_D);
  float4* d = (float4*)(out + (size_t)b * OUT_D);
  d[threadIdx.x] = s[threadIdx.x];  // 128 * 16B = 512 fp32
}

extern "C" void kernel_launch(void* const* d_in, const int* in_sizes, int n_in,
                              void* d_out, int out_size, void* d_ws, size_t ws_size,
                              hipStream_t stream) {
  (void)in_sizes; (void)n_in; (void)out_size; (void)ws_size;
  const int* tokens      = (const int*)d_in[0];
  const int* eidx        = (const int*)d_in[1];
  const float* edge_attr = (const float*)d_in[2];
  const int* batch       = (const int*)d_in[3];
  const float* emb       = (const float*)d_in[4];
  const float* proj_w    = (const float*)d_in[5];
  const float* proj_b    = (const float*)d_in[6];
  const int* src = eidx;           // edge_index[0,:]
  const int* dst = eidx + EDGES;   // edge_index[1,:]

  // workspace layout (fp32), total ~135.2M floats (~541 MB)
  float* ws  = (float*)d_ws;
  float* xA  = ws;                               // [N,512] ping
  float* xB  = xA  + (size_t)NODES * 512;        // [N,512] pong
  float* ea  = xB  + (size_t)NODES * 512;        // [E,48] current edge feats
  float* ewb = ea  + (size_t)EDGES * EPD_D;      // [E,48] ea@We + be scratch
  float* ms  = ewb + (size_t)EDGES * EPD_D;      // [N,48] x@Ws
  float* md  = ms  + (size_t)NODES * EPD_D;      // [N,48] x@Wd
  float* agg = md  + (size_t)NODES * EPD_D;      // [N,48] segment sum

  auto gemm = [&](const float* A, const float* W, const float* bias, float* C,
                  int M, int Nn, int K, int acc, int relu) {
    dim3 grid(M / 16, (Nn + 63) / 64);
    k_gemm_wmma<<<grid, 128, 0, stream>>>(A, W, bias, C, M, Nn, K, acc, relu);
  };

  // 1) tokens -> embeddings
  k_embed<<<NODES, 128, 0, stream>>>(tokens, emb, xA);
  // 2) project edges: ea = edge_attr @ proj_w + proj_b   [E,16]x[16,48]
  gemm(edge_attr, proj_w, proj_b, ea, EDGES, EPD_D, 16, 0, 0);

  static const int dims[4] = {EMB_D, MID_D, MID_D, OUT_D};
  float* xc = xA;
  float* xn = xB;
  for (int l = 0; l < 3; ++l) {
    const float* Ws = (const float*)d_in[7 + l * 7 + 0];
    const float* Wd = (const float*)d_in[7 + l * 7 + 1];
    const float* We = (const float*)d_in[7 + l * 7 + 2];
    const float* be = (const float*)d_in[7 + l * 7 + 3];
    const float* Wn = (const float*)d_in[7 + l * 7 + 4];
    const float* Wa = (const float*)d_in[7 + l * 7 + 5];
    const float* bn = (const float*)d_in[7 + l * 7 + 6];
    const int din = dims[l], dout = dims[l + 1];
    const int rl = (l < 2) ? 1 : 0;

    // project nodes BEFORE gathering (reference's cheap path)
    gemm(xc, Ws, nullptr, ms, NODES, EPD_D, din, 0, 0);
    gemm(xc, Wd, nullptr, md, NODES, EPD_D, din, 0, 0);
    gemm(ea, We, be, ewb, EDGES, EPD_D, EPD_D, 0, 0);

    hipMemsetAsync(agg, 0, (size_t)NODES * EPD_D * sizeof(float), stream);
    // e_new written in-place over `ea` (only layers 0,1 need it again)
    k_edge_combine<<<(EDGES + 255) / 256, 256, 0, stream>>>(
        src, dst, ms, md, ewb, ea, agg, rl, rl);

    // x_new = x@Wn + bn  ;  then  += agg@Wa  (+relu for layers 0,1)
    gemm(xc, Wn, bn, xn, NODES, dout, din, 0, 0);
    gemm(agg, Wa, nullptr, xn, NODES, dout, EPD_D, 1, rl);

    float* t = xc; xc = xn; xn = t;
  }

  // 3) master-node pool -> [B, 512]
  k_pool<<<GRAPHS, 128, 0, stream>>>(xc, batch, (float*)d_out);
}